// AffordanceClassifier_70549132804323
// MI455X (gfx1250) — compile-verified
//
#include <hip/hip_runtime.h>
#include <hip/hip_bf16.h>

typedef __attribute__((ext_vector_type(2))) float v2f;
typedef __attribute__((ext_vector_type(8))) float v8f;

#define NROWS 147456   // 1024*8*18
#define PC    144      // 8*18
#define BATCH 1024
#define D0    300
#define F1    128
#define F2    64
#define EPSBN 1e-5f

// ---------------------------------------------------------------------------
// Kernel 1: y1 = zs @ W1 + b1      [147456 x 300] * [300 x 128]
// Block: 256 thr = 8 waves; block tile = 128 rows x 128 cols; wave = 16 rows.
// W1 staged in LDS as K-pairs (float2 = {W1[2p][c], W1[2p+1][c]}) in 3 chunks
// of 100 K. B-fragment: lane half h reads pair (2s+h) -> one 8B LDS load.
// ---------------------------------------------------------------------------
#define G1_PAIRS 50          // pairs per chunk (100 K)
#define G1_LDSTR 130         // padded row stride in float2 (breaks bank conflict)

__global__ __launch_bounds__(256) void gemm1_kernel(
    const float* __restrict__ zs, const float* __restrict__ W1,
    const float* __restrict__ b1, float* __restrict__ y1) {
  __shared__ float2 w1s[G1_PAIRS * G1_LDSTR];   // 52,000 B

  const int tid  = threadIdx.x;
  const int lane = tid & 31;
  const int wave = tid >> 5;
  const int half = lane >> 4;     // selects K+0/1 vs K+2/3
  const int l16  = lane & 15;
  const int rowBase = blockIdx.x * 128 + wave * 16;
  const int aRow    = rowBase + l16;

  v8f acc[8];
  const v8f vzero = {0.f,0.f,0.f,0.f,0.f,0.f,0.f,0.f};
  #pragma unroll
  for (int t = 0; t < 8; ++t) acc[t] = vzero;

  for (int ch = 0; ch < 3; ++ch) {
    __syncthreads();
    // stage 100 K-rows of W1 as 50 pairs x 128 cols
    for (int i = tid; i < G1_PAIRS * F1; i += 256) {
      const int p = i >> 7, c = i & 127;
      const int kg = ch * 100 + 2 * p;
      w1s[p * G1_LDSTR + c] =
          make_float2(W1[kg * F1 + c], W1[(kg + 1) * F1 + c]);
    }
    __syncthreads();

    for (int s = 0; s < 25; ++s) {
      const int kbase = ch * 100 + 4 * s + 2 * half;
      // A fragment: 16x4 f32 layout (lane=M, half=K hi/lo pair)
      const float2 av = *(const float2*)&zs[(size_t)aRow * D0 + kbase];
      v2f a; a.x = av.x; a.y = av.y;

      const float2* brow = &w1s[(2 * s + half) * G1_LDSTR];
      #pragma unroll
      for (int t = 0; t < 8; ++t) {
        const float2 bv = brow[t * 16 + l16];
        v2f b; b.x = bv.x; b.y = bv.y;
        acc[t] = __builtin_amdgcn_wmma_f32_16x16x4_f32(
            false, a, false, b, (short)0, acc[t], false, false);
      }
    }
  }

  // C/D layout: VGPR r -> row r (lanes 0-15) / r+8 (lanes 16-31), col = l16
  #pragma unroll
  for (int t = 0; t < 8; ++t) {
    const int col  = t * 16 + l16;
    const float bb = b1[col];
    #pragma unroll
    for (int r = 0; r < 8; ++r) {
      const int row = rowBase + r + half * 8;
      y1[(size_t)row * F1 + col] = acc[t][r] + bb;
    }
  }
}

// ---------------------------------------------------------------------------
// Kernel 2: BN1 stats over batch per (pc, f); fold gamma/beta into scale/shift
// Block = one pc slice [1024 x 128]; 1024 thr = 128 f x 8 batch-groups.
// ---------------------------------------------------------------------------
__global__ __launch_bounds__(1024) void reduce1_kernel(
    const float* __restrict__ y1, const float* __restrict__ g1,
    const float* __restrict__ be1, float* __restrict__ scale1,
    float* __restrict__ shift1) {
  __shared__ float ssum[1024];
  __shared__ float ssq[1024];
  const int pc = blockIdx.x;
  const int f  = threadIdx.x & 127;
  const int bg = threadIdx.x >> 7;

  float s = 0.f, q = 0.f;
  for (int b = bg * 128; b < bg * 128 + 128; ++b) {
    const float v = y1[(size_t)(b * PC + pc) * F1 + f];
    s += v; q += v * v;
  }
  ssum[threadIdx.x] = s; ssq[threadIdx.x] = q;
  __syncthreads();
  if (bg == 0) {
    #pragma unroll
    for (int g = 1; g < 8; ++g) { s += ssum[g * 128 + f]; q += ssq[g * 128 + f]; }
    const float mean = s * (1.f / BATCH);
    const float var  = q * (1.f / BATCH) - mean * mean;
    const float sc   = g1[f] * rsqrtf(var + EPSBN);
    scale1[pc * F1 + f] = sc;
    shift1[pc * F1 + f] = be1[f] - mean * sc;
  }
}

// ---------------------------------------------------------------------------
// Kernel 3: y2 = relu(y1*scale1+shift1) @ W2 + b2   [N x 128] * [128 x 64]
// BN1 + ReLU fused into the A-fragment load. W2 fully LDS-resident as pairs.
// ---------------------------------------------------------------------------
#define G2_LDSTR 66

__global__ __launch_bounds__(256) void gemm2_kernel(
    const float* __restrict__ y1, const float* __restrict__ scale1,
    const float* __restrict__ shift1, const float* __restrict__ W2,
    const float* __restrict__ b2, float* __restrict__ y2) {
  __shared__ float2 w2s[64 * G2_LDSTR];   // 33,792 B

  const int tid  = threadIdx.x;
  const int lane = tid & 31;
  const int wave = tid >> 5;
  const int half = lane >> 4;
  const int l16  = lane & 15;
  const int rowBase = blockIdx.x * 128 + wave * 16;
  const int aRow    = rowBase + l16;
  const int pc      = aRow % PC;

  for (int i = tid; i < 64 * F2; i += 256) {
    const int p = i >> 6, c = i & 63;
    w2s[p * G2_LDSTR + c] =
        make_float2(W2[(2 * p) * F2 + c], W2[(2 * p + 1) * F2 + c]);
  }
  __syncthreads();

  v8f acc[4];
  const v8f vzero = {0.f,0.f,0.f,0.f,0.f,0.f,0.f,0.f};
  #pragma unroll
  for (int t = 0; t < 4; ++t) acc[t] = vzero;

  const float* scp = &scale1[pc * F1];
  const float* shp = &shift1[pc * F1];

  for (int s = 0; s < 32; ++s) {
    const int kbase = 4 * s + 2 * half;
    const float2 av  = *(const float2*)&y1[(size_t)aRow * F1 + kbase];
    const float2 scv = *(const float2*)&scp[kbase];
    const float2 shv = *(const float2*)&shp[kbase];
    v2f a;
    a.x = fmaxf(av.x * scv.x + shv.x, 0.f);   // fused BN1 + ReLU
    a.y = fmaxf(av.y * scv.y + shv.y, 0.f);

    const float2* brow = &w2s[(2 * s + half) * G2_LDSTR];
    #pragma unroll
    for (int t = 0; t < 4; ++t) {
      const float2 bv = brow[t * 16 + l16];
      v2f b; b.x = bv.x; b.y = bv.y;
      acc[t] = __builtin_amdgcn_wmma_f32_16x16x4_f32(
          false, a, false, b, (short)0, acc[t], false, false);
    }
  }

  #pragma unroll
  for (int t = 0; t < 4; ++t) {
    const int col  = t * 16 + l16;
    const float bb = b2[col];
    #pragma unroll
    for (int r = 0; r < 8; ++r) {
      const int row = rowBase + r + half * 8;
      y2[(size_t)row * F2 + col] = acc[t][r] + bb;
    }
  }
}

// ---------------------------------------------------------------------------
// Kernel 4: BN2 stats; 1024 thr = 64 f x 16 batch-groups per pc slice.
// ---------------------------------------------------------------------------
__global__ __launch_bounds__(1024) void reduce2_kernel(
    const float* __restrict__ y2, const float* __restrict__ g2,
    const float* __restrict__ be2, float* __restrict__ scale2,
    float* __restrict__ shift2) {
  __shared__ float ssum[1024];
  __shared__ float ssq[1024];
  const int pc = blockIdx.x;
  const int f  = threadIdx.x & 63;
  const int bg = threadIdx.x >> 6;

  float s = 0.f, q = 0.f;
  for (int b = bg * 64; b < bg * 64 + 64; ++b) {
    const float v = y2[(size_t)(b * PC + pc) * F2 + f];
    s += v; q += v * v;
  }
  ssum[threadIdx.x] = s; ssq[threadIdx.x] = q;
  __syncthreads();
  if (bg == 0) {
    #pragma unroll
    for (int g = 1; g < 16; ++g) { s += ssum[g * 64 + f]; q += ssq[g * 64 + f]; }
    const float mean = s * (1.f / BATCH);
    const float var  = q * (1.f / BATCH) - mean * mean;
    const float sc   = g2[f] * rsqrtf(var + EPSBN);
    scale2[pc * F2 + f] = sc;
    shift2[pc * F2 + f] = be2[f] - mean * sc;
  }
}

// ---------------------------------------------------------------------------
// Kernel 5: out = sigmoid(relu(y2*scale2+shift2) @ W3 + b3). One thread/row.
// ---------------------------------------------------------------------------
__global__ __launch_bounds__(256) void final_kernel(
    const float* __restrict__ y2, const float* __restrict__ scale2,
    const float* __restrict__ shift2, const float* __restrict__ W3,
    const float* __restrict__ b3, float* __restrict__ out) {
  const int row = blockIdx.x * 256 + threadIdx.x;
  const int pc  = row % PC;
  const float4* yv  = (const float4*)&y2[(size_t)row * F2];
  const float4* scv = (const float4*)&scale2[pc * F2];
  const float4* shv = (const float4*)&shift2[pc * F2];
  const float4* wv  = (const float4*)W3;

  float accum = 0.f;
  #pragma unroll
  for (int i = 0; i < 16; ++i) {
    const float4 y = yv[i], sc = scv[i], sh = shv[i], w = wv[i];
    accum += fmaxf(y.x * sc.x + sh.x, 0.f) * w.x;
    accum += fmaxf(y.y * sc.y + sh.y, 0.f) * w.y;
    accum += fmaxf(y.z * sc.z + sh.z, 0.f) * w.z;
    accum += fmaxf(y.w * sc.w + sh.w, 0.f) * w.w;
  }
  const float logit = accum + b3[0];
  out[row] = 1.f / (1.f + expf(-logit));
}

// ---------------------------------------------------------------------------
extern "C" void kernel_launch(void* const* d_in, const int* in_sizes, int n_in,
                              void* d_out, int out_size, void* d_ws, size_t ws_size,
                              hipStream_t stream) {
  const float* zs  = (const float*)d_in[0];
  const float* W1  = (const float*)d_in[1];
  const float* b1  = (const float*)d_in[2];
  const float* g1  = (const float*)d_in[3];
  const float* be1 = (const float*)d_in[4];
  const float* W2  = (const float*)d_in[5];
  const float* b2  = (const float*)d_in[6];
  const float* g2  = (const float*)d_in[7];
  const float* be2 = (const float*)d_in[8];
  const float* W3  = (const float*)d_in[9];
  const float* b3  = (const float*)d_in[10];
  float* out = (float*)d_out;

  char* ws = (char*)d_ws;
  float* y1     = (float*)ws;                                   // 75,497,472 B
  float* y2     = (float*)(ws + (size_t)NROWS * F1 * 4);        // 37,748,736 B
  float* scale1 = (float*)(ws + (size_t)NROWS * (F1 + F2) * 4); // small tail
  float* shift1 = scale1 + PC * F1;
  float* scale2 = shift1 + PC * F1;
  float* shift2 = scale2 + PC * F2;

  gemm1_kernel  <<<NROWS / 128, 256,  0, stream>>>(zs, W1, b1, y1);
  reduce1_kernel<<<PC,          1024, 0, stream>>>(y1, g1, be1, scale1, shift1);
  gemm2_kernel  <<<NROWS / 128, 256,  0, stream>>>(y1, scale1, shift1, W2, b2, y2);
  reduce2_kernel<<<PC,          1024, 0, stream>>>(y2, g2, be2, scale2, shift2);
  final_kernel  <<<NROWS / 256, 256,  0, stream>>>(y2, scale2, shift2, W3, b3, out);
}